// NegativeSamplingPyTorch_90254442758236
// MI455X (gfx1250) — compile-verified
//
#include <hip/hip_runtime.h>

typedef float v2f __attribute__((ext_vector_type(2)));
typedef float v8f __attribute__((ext_vector_type(8)));

#define W2V_VOCAB 100000
#define W2V_DIM   128
#define W2V_B     262144
#define W2V_NEG   5

#define NBLOCKS 1024
#define TPB     256
#define WAVES_PER_BLOCK (TPB / 32)

// stable softplus(x) = log(1 + e^x) = max(x,0) + log(1 + e^{-|x|})
__device__ __forceinline__ float softplus_f(float x) {
    float e = __expf(-fabsf(x));
    return fmaxf(x, 0.0f) + __logf(1.0f + e);
}

// pick element i (0..7) of a v8f with a cndmask chain (no EXEC divergence)
__device__ __forceinline__ float pick8(v8f v, int i) {
    float r = v[0];
    r = (i == 1) ? v[1] : r;
    r = (i == 2) ? v[2] : r;
    r = (i == 3) ? v[3] : r;
    r = (i == 4) ? v[4] : r;
    r = (i == 5) ? v[5] : r;
    r = (i == 6) ? v[6] : r;
    r = (i == 7) ? v[7] : r;
    return r;
}

__global__ __launch_bounds__(TPB)
void w2v_neg_sampling_wmma(const int* __restrict__ tgt,
                           const int* __restrict__ ctx,
                           const int* __restrict__ neg,
                           const float* __restrict__ iemb,
                           const float* __restrict__ oemb,
                           float* __restrict__ ws) {
    const int lane = threadIdx.x & 31;
    const int wid  = threadIdx.x >> 5;
    const int row  = lane & 15;      // sample row within 16-sample group
    const int koff = (lane >> 4) * 2; // lane K sub-offset: 0 or 2 (A and B layouts match)

    const int gwave   = blockIdx.x * WAVES_PER_BLOCK + wid;
    const int nwaves  = gridDim.x * WAVES_PER_BLOCK;
    const int ngroups = W2V_B / 16;

    float pos_acc = 0.0f;
    float neg_acc = 0.0f;

    for (int g = gwave; g < ngroups; g += nwaves) {   // uniform per wave -> EXEC stays full
        const int s = g * 16 + row;

        // gathered row base pointers, pre-offset by the lane's K sub-column
        const float* tptr  = iemb + (size_t)tgt[s] * W2V_DIM + koff;
        const float* cptr  = oemb + (size_t)ctx[s] * W2V_DIM + koff;
        const float* nptr0 = oemb + (size_t)neg[s * W2V_NEG + 0] * W2V_DIM + koff;
        const float* nptr1 = oemb + (size_t)neg[s * W2V_NEG + 1] * W2V_DIM + koff;
        const float* nptr2 = oemb + (size_t)neg[s * W2V_NEG + 2] * W2V_DIM + koff;
        const float* nptr3 = oemb + (size_t)neg[s * W2V_NEG + 3] * W2V_DIM + koff;
        const float* nptr4 = oemb + (size_t)neg[s * W2V_NEG + 4] * W2V_DIM + koff;

        v8f accP = {};
        v8f accN0 = {}, accN1 = {}, accN2 = {}, accN3 = {}, accN4 = {};

        #pragma unroll
        for (int k = 0; k < W2V_DIM; k += 4) {
            v2f a  = *(const v2f*)(tptr  + k);
            v2f bc = *(const v2f*)(cptr  + k);
            v2f b0 = *(const v2f*)(nptr0 + k);
            v2f b1 = *(const v2f*)(nptr1 + k);
            v2f b2 = *(const v2f*)(nptr2 + k);
            v2f b3 = *(const v2f*)(nptr3 + k);
            v2f b4 = *(const v2f*)(nptr4 + k);
            // D = A x B + C, f32 16x16x4 (8-arg form: neg_a, A, neg_b, B, c_mod, C, reuse_a, reuse_b)
            accP  = __builtin_amdgcn_wmma_f32_16x16x4_f32(false, a, false, bc, (short)0, accP,  false, false);
            accN0 = __builtin_amdgcn_wmma_f32_16x16x4_f32(false, a, false, b0, (short)0, accN0, false, false);
            accN1 = __builtin_amdgcn_wmma_f32_16x16x4_f32(false, a, false, b1, (short)0, accN1, false, false);
            accN2 = __builtin_amdgcn_wmma_f32_16x16x4_f32(false, a, false, b2, (short)0, accN2, false, false);
            accN3 = __builtin_amdgcn_wmma_f32_16x16x4_f32(false, a, false, b3, (short)0, accN3, false, false);
            accN4 = __builtin_amdgcn_wmma_f32_16x16x4_f32(false, a, false, b4, (short)0, accN4, false, false);
        }

        // Diagonal of D lives at: lanes 0..7 -> element=lane (m=lane);
        //                         lanes 24..31 -> element=lane-24 (m=lane-16).
        const bool has = (lane < 8) || (lane >= 24);
        const int  di  = (lane < 8) ? lane : (lane - 24);

        float sp = pick8(accP,  di);
        float s0 = pick8(accN0, di);
        float s1 = pick8(accN1, di);
        float s2 = pick8(accN2, di);
        float s3 = pick8(accN3, di);
        float s4 = pick8(accN4, di);

        float lp = softplus_f(-sp);                    // -log_sigmoid(score)
        float ln = softplus_f(s0) + softplus_f(s1) + softplus_f(s2)
                 + softplus_f(s3) + softplus_f(s4);    // -log_sigmoid(-score)

        pos_acc += has ? lp : 0.0f;
        neg_acc += has ? ln : 0.0f;
    }

    // deterministic fixed-order wave32 reduction
    #pragma unroll
    for (int off = 16; off > 0; off >>= 1) {
        pos_acc += __shfl_down(pos_acc, off, 32);
        neg_acc += __shfl_down(neg_acc, off, 32);
    }

    __shared__ float wpos[WAVES_PER_BLOCK];
    __shared__ float wneg[WAVES_PER_BLOCK];
    if (lane == 0) { wpos[wid] = pos_acc; wneg[wid] = neg_acc; }
    __syncthreads();
    if (threadIdx.x == 0) {
        float p = 0.0f, n = 0.0f;
        #pragma unroll
        for (int i = 0; i < WAVES_PER_BLOCK; ++i) { p += wpos[i]; n += wneg[i]; }
        ws[2 * blockIdx.x + 0] = p;
        ws[2 * blockIdx.x + 1] = n;
    }
}

__global__ __launch_bounds__(256)
void w2v_finalize(const float* __restrict__ ws, float* __restrict__ out) {
    __shared__ float sp[256];
    __shared__ float sn[256];
    const int t = threadIdx.x;
    float p = 0.0f, n = 0.0f;
    for (int i = t; i < NBLOCKS; i += 256) {  // fixed order per thread
        p += ws[2 * i + 0];
        n += ws[2 * i + 1];
    }
    sp[t] = p; sn[t] = n;
    __syncthreads();
    for (int s = 128; s > 0; s >>= 1) {       // fixed pairwise tree
        if (t < s) { sp[t] += sp[t + s]; sn[t] += sn[t + s]; }
        __syncthreads();
    }
    if (t == 0) {
        out[0] = sp[0] * (1.0f / (float)W2V_B);
        out[1] = sn[0] * (1.0f / ((float)W2V_B * (float)W2V_NEG));
    }
}

extern "C" void kernel_launch(void* const* d_in, const int* in_sizes, int n_in,
                              void* d_out, int out_size, void* d_ws, size_t ws_size,
                              hipStream_t stream) {
    (void)in_sizes; (void)n_in; (void)out_size; (void)ws_size;
    const int*   tgt  = (const int*)d_in[0];
    const int*   ctx  = (const int*)d_in[1];
    const int*   neg  = (const int*)d_in[2];
    const float* iemb = (const float*)d_in[3];
    const float* oemb = (const float*)d_in[4];
    float* out = (float*)d_out;
    float* ws  = (float*)d_ws;

    w2v_neg_sampling_wmma<<<NBLOCKS, TPB, 0, stream>>>(tgt, ctx, neg, iemb, oemb, ws);
    w2v_finalize<<<1, 256, 0, stream>>>(ws, out);
}